// EcTransOctahedralSHT_82300163326242
// MI455X (gfx1250) — compile-verified
//
#include <hip/hip_runtime.h>

typedef __attribute__((ext_vector_type(2))) float v2f;
typedef __attribute__((ext_vector_type(4))) float v4f;
typedef __attribute__((ext_vector_type(8))) float v8f;

#define NLAT    128   // N_LAT_NH
#define NRINGS  256
#define MMAX    128
#define NVARS   128
#define NBE     2     // BS * ENS
#define MAXLONS 528
#define NGRID   70144
#define NSYM    64    // n_sym == n_anti

static __device__ __forceinline__ v8f wmma4(v2f a, v2f b, v8f c) {
    return __builtin_amdgcn_wmma_f32_16x16x4_f32(false, a, false, b,
                                                 (short)0, c, false, false);
}

// ring start offset: cum[l] = 2l^2+18l (north); n_grid - (2t^2+18t), t=256-l (south).
// Identical to ring_idx[l,j]-j for all j < lons[l] (clamp only hits the padded tail).
static __device__ __forceinline__ int ring_start(int l) {
    if (l < NLAT) return 2 * l * l + 18 * l;
    const int t = NRINGS - l;
    return NGRID - (2 * t * t + 18 * t);
}

// ---------------------------------------------------------------------------
// Stage 1: per-ring DFT. F[be,m,l,v] = sum_j W[l,m,j] * x[be, cum[l]+j, v]
// One wave: 16(m) x 32(v) complex tile, j-loop unrolled x2 (K=8 per iter).
// Tile decode goes through readfirstlane so loop bounds are scalar and all
// branches are s_cbranch (EXEC stays all-ones around the WMMAs).
// F stored planar re/im with layout [be][m][ring][v] for stage-2 locality.
// ---------------------------------------------------------------------------
__global__ __launch_bounds__(256) void sht_dft_stage1(
    const float* __restrict__ x,   // (NBE, NGRID, NVARS) f32
    const float* __restrict__ W,   // (NRINGS, MMAX, MAXLONS) complex64 interleaved
    float*       __restrict__ Fre, // (NBE, MMAX, NRINGS, NVARS)
    float*       __restrict__ Fim)
{
    const int lane = threadIdx.x & 31;
    const int wid  = threadIdx.x >> 5;
    // Wave-uniform tile index, made explicit for the compiler (scalar branches).
    const unsigned g =
        (unsigned)__builtin_amdgcn_readfirstlane((int)(blockIdx.x * 8u + wid));

    const int vt = g & 3;          // 32-wide v tile 0..3
    const int mt = (g >> 2) & 7;   // m tile 0..7
    const int be = (g >> 5) & 1;   // batch
    const int l  = (int)(g >> 6);  // ring 0..255

    const int lons = 20 + 4 * (l < NLAT ? l : (NRINGS - 1 - l));  // SGPR
    const int m0 = mt * 16;
    const int v0 = vt * 32;

    const int half = lane >> 4;    // K half: 0 -> K{0,1}, 1 -> K{2,3}
    const int lq   = lane & 15;

    // A source: this lane's m-row of W, starting at complex j = 2*half.
    const float* wp = W + (size_t)(l * MMAX + m0 + lq) * (MAXLONS * 2) + 4 * half;
    // B source: contiguous x rows starting at ring_start(l) + 2*half.
    const float* xp = x + ((size_t)be * NGRID + ring_start(l) + 2 * half) * NVARS
                    + v0 + lq;

    v8f cre0 = {}, cim0 = {}, cre1 = {}, cim1 = {};

    int j = 0;
    if (lons & 4) {                       // scalar condition -> s_cbranch
        v4f wv = *(const v4f*)wp;         // {re0, im0, re1, im1}
        v2f are; are.x = wv.x; are.y = wv.z;
        v2f aim; aim.x = wv.y; aim.y = wv.w;
        v2f b0;  b0.x = xp[0];       b0.y = xp[NVARS];
        v2f b1;  b1.x = xp[16];      b1.y = xp[NVARS + 16];
        cre0 = wmma4(are, b0, cre0);  cim0 = wmma4(aim, b0, cim0);
        cre1 = wmma4(are, b1, cre1);  cim1 = wmma4(aim, b1, cim1);
        wp += 8;
        xp += 4 * NVARS;
        j = 4;
    }
    for (; j < lons; j += 8) {            // scalar trip count -> s_cbranch loop
        v4f wv0 = *(const v4f*)wp;
        v4f wv1 = *(const v4f*)(wp + 8);
        v2f are0; are0.x = wv0.x; are0.y = wv0.z;
        v2f aim0; aim0.x = wv0.y; aim0.y = wv0.w;
        v2f are1; are1.x = wv1.x; are1.y = wv1.z;
        v2f aim1; aim1.x = wv1.y; aim1.y = wv1.w;

        v2f b00; b00.x = xp[0];              b00.y = xp[NVARS];
        v2f b01; b01.x = xp[16];             b01.y = xp[NVARS + 16];
        v2f b10; b10.x = xp[4 * NVARS];      b10.y = xp[5 * NVARS];
        v2f b11; b11.x = xp[4 * NVARS + 16]; b11.y = xp[5 * NVARS + 16];

        cre0 = wmma4(are0, b00, cre0);  cim0 = wmma4(aim0, b00, cim0);
        cre1 = wmma4(are0, b01, cre1);  cim1 = wmma4(aim0, b01, cim1);
        cre0 = wmma4(are1, b10, cre0);  cim0 = wmma4(aim1, b10, cim0);
        cre1 = wmma4(are1, b11, cre1);  cim1 = wmma4(aim1, b11, cim1);

        wp += 16;
        xp += 8 * NVARS;
    }

    // C/D layout: lanes 0-15 -> M = r, lanes 16-31 -> M = r + 8; N = lane & 15.
    const size_t mstride = (size_t)NRINGS * NVARS;
    float* pre = Fre + (((size_t)be * MMAX + m0) * NRINGS + l) * NVARS + v0 + lq;
    float* pim = Fim + (((size_t)be * MMAX + m0) * NRINGS + l) * NVARS + v0 + lq;
#pragma unroll
    for (int r = 0; r < 8; ++r) {
        const size_t off = (size_t)(r + 8 * half) * mstride;
        pre[off]      = cre0[r];
        pre[off + 16] = cre1[r];
        pim[off]      = cim0[r];
        pim[off + 16] = cim1[r];
    }
}

// ---------------------------------------------------------------------------
// Stage 2: Legendre contraction, both parities fused (they share all F loads):
//   odd : spec[2l+1,m,v] = sum_i sym [m,l,i] * (F[i,m,v] + F[255-i,m,v])
//   even: spec[2l  ,m,v] = sum_i anti[m,l,i] * (F[i,m,v] - F[255-i,m,v])
// One wave: 16(l) x 32(v), 8 accumulators, 8 WMMAs per K=4 step.
// ---------------------------------------------------------------------------
__global__ __launch_bounds__(256) void sht_leg_stage2(
    const float* __restrict__ sym,   // (MMAX, NSYM, NLAT) complex64 (imag = 0)
    const float* __restrict__ anti,  // (MMAX, NSYM, NLAT) complex64 (imag = 0)
    const float* __restrict__ Fre,   // (NBE, MMAX, NRINGS, NVARS)
    const float* __restrict__ Fim,
    float*       __restrict__ out)   // (NBE, T+1, MMAX, NVARS) complex64 interleaved
{
    const int lane = threadIdx.x & 31;
    const int wid  = threadIdx.x >> 5;
    const unsigned g =
        (unsigned)__builtin_amdgcn_readfirstlane((int)(blockIdx.x * 8u + wid));

    const int vt = g & 3;            // 32-wide v tile 0..3
    const int lt = (g >> 2) & 3;     // l tile 0..3
    const int m  = (g >> 4) & 127;   // m 0..127
    const int be = (int)(g >> 11);   // batch

    const int l0 = lt * 16;
    const int v0 = vt * 32;
    const int half = lane >> 4;
    const int lq   = lane & 15;

    const float* As = sym  + (size_t)(m * NSYM + l0 + lq) * (NLAT * 2) + 4 * half;
    const float* Aa = anti + (size_t)(m * NSYM + l0 + lq) * (NLAT * 2) + 4 * half;

    const float* fre = Fre + ((size_t)be * MMAX + m) * ((size_t)NRINGS * NVARS)
                     + v0 + lq;
    const float* fim = Fim + ((size_t)be * MMAX + m) * ((size_t)NRINGS * NVARS)
                     + v0 + lq;

    v8f csr0 = {}, csi0 = {}, car0 = {}, cai0 = {};
    v8f csr1 = {}, csi1 = {}, car1 = {}, cai1 = {};

#pragma unroll 2
    for (int i = 0; i < NLAT; i += 4) {
        const int k0 = i + 2 * half;

        v4f sv = *(const v4f*)(As + 2 * i);     // coef[m,l,k0..k0+1] as complex
        v4f av = *(const v4f*)(Aa + 2 * i);
        v2f as; as.x = sv.x; as.y = sv.z;       // real parts only (imag == 0)
        v2f aa; aa.x = av.x; aa.y = av.z;

        const size_t n0 = (size_t)k0 * NVARS;
        const size_t n1 = (size_t)(k0 + 1) * NVARS;
        const size_t s0 = (size_t)(NRINGS - 1 - k0) * NVARS;
        const size_t s1 = (size_t)(NRINGS - 2 - k0) * NVARS;

        // v-tile 0
        float nr0 = fre[n0], nr1 = fre[n1], sr0 = fre[s0], sr1 = fre[s1];
        float ni0 = fim[n0], ni1 = fim[n1], si0 = fim[s0], si1 = fim[s1];
        v2f bsr; bsr.x = nr0 + sr0; bsr.y = nr1 + sr1;
        v2f bar; bar.x = nr0 - sr0; bar.y = nr1 - sr1;
        v2f bsi; bsi.x = ni0 + si0; bsi.y = ni1 + si1;
        v2f bai; bai.x = ni0 - si0; bai.y = ni1 - si1;
        csr0 = wmma4(as, bsr, csr0);  csi0 = wmma4(as, bsi, csi0);
        car0 = wmma4(aa, bar, car0);  cai0 = wmma4(aa, bai, cai0);

        // v-tile 1 (+16)
        nr0 = fre[n0 + 16]; nr1 = fre[n1 + 16]; sr0 = fre[s0 + 16]; sr1 = fre[s1 + 16];
        ni0 = fim[n0 + 16]; ni1 = fim[n1 + 16]; si0 = fim[s0 + 16]; si1 = fim[s1 + 16];
        bsr.x = nr0 + sr0; bsr.y = nr1 + sr1;
        bar.x = nr0 - sr0; bar.y = nr1 - sr1;
        bsi.x = ni0 + si0; bsi.y = ni1 + si1;
        bai.x = ni0 - si0; bai.y = ni1 - si1;
        csr1 = wmma4(as, bsr, csr1);  csi1 = wmma4(as, bsi, csi1);
        car1 = wmma4(aa, bar, car1);  cai1 = wmma4(aa, bai, cai1);
    }

    // Store interleaved complex pairs as packed b64; odd rows from sym, even from anti.
#pragma unroll
    for (int r = 0; r < 8; ++r) {
        const int lrow = l0 + r + 8 * half;          // 0..63
        const size_t iodd  = (((size_t)be * 128 + (2 * lrow + 1)) * MMAX + m) * NVARS
                           + v0 + lq;
        const size_t ieven = (((size_t)be * 128 + (2 * lrow)) * MMAX + m) * NVARS
                           + v0 + lq;
        v2f o;
        o.x = csr0[r]; o.y = csi0[r]; *(v2f*)(out + 2 * iodd)         = o;
        o.x = car0[r]; o.y = cai0[r]; *(v2f*)(out + 2 * ieven)        = o;
        o.x = csr1[r]; o.y = csi1[r]; *(v2f*)(out + 2 * (iodd + 16))  = o;
        o.x = car1[r]; o.y = cai1[r]; *(v2f*)(out + 2 * (ieven + 16)) = o;
    }
}

// ---------------------------------------------------------------------------
extern "C" void kernel_launch(void* const* d_in, const int* in_sizes, int n_in,
                              void* d_out, int out_size, void* d_ws, size_t ws_size,
                              hipStream_t stream) {
    const float* x    = (const float*)d_in[0];  // (2,1,70144,128) f32
    const float* W    = (const float*)d_in[1];  // (256,128,528) c64
    const float* sym  = (const float*)d_in[3];  // (128,64,128) c64
    const float* anti = (const float*)d_in[4];  // (128,64,128) c64
    float*       out  = (float*)d_out;          // (2,128,128,128) c64

    const size_t fplane = (size_t)NBE * MMAX * NRINGS * NVARS;  // 8.39M floats
    float* Fre = (float*)d_ws;
    float* Fim = Fre + fplane;

    // Stage 1: 256 l * 2 be * 8 mt * 4 vt = 16384 waves, 8 per block.
    sht_dft_stage1<<<2048, 256, 0, stream>>>(x, W, Fre, Fim);
    // Stage 2: 2 be * 128 m * 4 lt * 4 vt = 4096 waves.
    sht_leg_stage2<<<512, 256, 0, stream>>>(sym, anti, Fre, Fim, out);
}